// NeighborRoutingAgg_65025804861640
// MI455X (gfx1250) — compile-verified
//
#include <hip/hip_runtime.h>

// NeighborRoutingAgg on MI455X (gfx1250, wave32)
//
// Strategy: one wave per node. Gather the 32x128 neighbor tile of the
// pre-normalized x into LDS once (L2-resident gather is the real cost:
// ~820MB of L2 traffic vs ~60MB HBM). Reformulate the 3 routing rounds via
// the Gram matrix G = Z Z^T (computed with fp32 WMMA 16x16x4, 100% useful
// FLOPs) so that logits = s*(G p + b) and ||u||^2 = p^T G p + 2 p^T b + 1,
// with b = Z.xn. Only one final Z^T p + xn pass over the tile.

#define D_DIM 128
#define M_NBR 32
#define ZSTR  132   // z-tile LDS row stride (floats): conflict-free WMMA operand reads
#define GSTR  36    // Gram LDS row stride (floats): conflict-free tile spill

typedef __attribute__((ext_vector_type(2))) float v2f;
typedef __attribute__((ext_vector_type(8))) float v8f;

static __device__ __forceinline__ float wave_sum32(float v) {
#pragma unroll
  for (int off = 16; off > 0; off >>= 1) v += __shfl_xor(v, off, 32);
  return v;
}
static __device__ __forceinline__ float wave_max32(float v) {
#pragma unroll
  for (int off = 16; off > 0; off >>= 1) v = fmaxf(v, __shfl_xor(v, off, 32));
  return v;
}

// Pass 1: xn = l2_normalize(x, axis=1). One wave per row, b128 per lane.
__global__ __launch_bounds__(256) void l2norm_rows(const float* __restrict__ x,
                                                   float* __restrict__ xn, int n) {
  int row  = blockIdx.x * 8 + (threadIdx.x >> 5);
  int lane = threadIdx.x & 31;
  if (row >= n) return;
  float4 v = ((const float4*)x)[(size_t)row * (D_DIM / 4) + lane];
  float s = v.x * v.x + v.y * v.y + v.z * v.z + v.w * v.w;
  s = wave_sum32(s);
  float sc = 1.0f / fmaxf(sqrtf(s), 1e-12f);
  float4 o; o.x = v.x * sc; o.y = v.y * sc; o.z = v.z * sc; o.w = v.w * sc;
  ((float4*)xn)[(size_t)row * (D_DIM / 4) + lane] = o;
}

// Pass 2: per-node routing. blockDim = 32 (single-wave workgroup).
__global__ __launch_bounds__(32) void capsule_route(const float* __restrict__ xn,
                                                    const int* __restrict__ x_nb,
                                                    float* __restrict__ out, int n) {
  __shared__ float zt[M_NBR][ZSTR];   // gathered neighbor tile (row-major, padded)
  __shared__ float xnrow[D_DIM];      // this node's xn row (broadcast reads)
  __shared__ float G[M_NBR][GSTR];    // Gram matrix Z Z^T
  __shared__ float pv[M_NBR];         // routing coefficients

  int node = blockIdx.x;
  if (node >= n) return;
  int L  = threadIdx.x;               // lane = neighbor slot
  int c4 = L * 4;

  int nbi = x_nb[(size_t)node * M_NBR + L] - 1;   // 0-based neighbor row

  // Cooperative gather: each of the 32 rows is one fully-coalesced 512B b128 load.
#pragma unroll 4
  for (int m = 0; m < M_NBR; ++m) {
    int r = __shfl(nbi, m, 32);
    float4 v = ((const float4*)xn)[(size_t)r * (D_DIM / 4) + L];
    *(float4*)&zt[m][c4] = v;
  }
  float4 xs = ((const float4*)xn)[(size_t)node * (D_DIM / 4) + L];
  *(float4*)&xnrow[c4] = xs;
  __syncthreads();

  // b[m] = <z_m, xn>  (lane m owns row m; xnrow reads are same-address broadcasts)
  float bm = 0.0f;
#pragma unroll 8
  for (int j = 0; j < D_DIM / 4; ++j) {
    float4 zv = *(const float4*)&zt[L][4 * j];
    float4 xv = *(const float4*)&xnrow[4 * j];
    bm += zv.x * xv.x + zv.y * xv.y + zv.z * xv.z + zv.w * xv.w;
  }

  // G = Z Z^T via V_WMMA_F32_16X16X4_F32.
  // fp32 A-operand layout: lane L holds A[L&15][2*(L>>4) + j] in reg j -> one ds b64.
  // For G = Z Z^T the B-operand per-lane layout coincides with A's, so the same
  // register pair serves both operands. 2 loads -> 4 independent MACC chains per K-step.
  int h = L >> 4, cc = L & 15;
  v8f g00 = {}, g01 = {}, g10 = {}, g11 = {};
#pragma unroll 4
  for (int k0 = 0; k0 < D_DIM; k0 += 4) {
    v2f a0 = *(const v2f*)&zt[cc][k0 + 2 * h];        // rows 0..15 block
    v2f a1 = *(const v2f*)&zt[16 + cc][k0 + 2 * h];   // rows 16..31 block
    g00 = __builtin_amdgcn_wmma_f32_16x16x4_f32(false, a0, false, a0, (short)0, g00, false, false);
    g01 = __builtin_amdgcn_wmma_f32_16x16x4_f32(false, a0, false, a1, (short)0, g01, false, false);
    g10 = __builtin_amdgcn_wmma_f32_16x16x4_f32(false, a1, false, a0, (short)0, g10, false, false);
    g11 = __builtin_amdgcn_wmma_f32_16x16x4_f32(false, a1, false, a1, (short)0, g11, false, false);
  }
  // Spill Gram (C/D layout: reg v holds D[v + 8*(L>>4)][L&15]).
#pragma unroll
  for (int v = 0; v < 8; ++v) {
    int r = v + 8 * h;
    G[r][cc]           = g00[v];
    G[r][16 + cc]      = g01[v];
    G[16 + r][cc]      = g10[v];
    G[16 + r][16 + cc] = g11[v];
  }
  __syncthreads();

  // Routing rounds in 32-d coefficient space (lane m <-> neighbor m).
  float p = 1.0f / 32.0f;   // softmax of zero logits
  for (int it = 0; it < 2; ++it) {
    pv[L] = p;
    __syncthreads();
    float gp = 0.0f;        // (G p)[L]
#pragma unroll
    for (int j = 0; j < 8; ++j) {
      float4 gr = *(const float4*)&G[L][4 * j];
      float4 pj = *(const float4*)&pv[4 * j];
      gp += gr.x * pj.x + gr.y * pj.y + gr.z * pj.z + gr.w * pj.w;
    }
    float sq = wave_sum32(p * (gp + 2.0f * bm)) + 1.0f;  // ||u||^2
    float s  = sqrtf(sq) / (sq + 1.0f);                  // squash scale on u
    float logit = s * (gp + bm);                         // Z.(s u)
    float mx = wave_max32(logit);
    float e  = __expf(logit - mx);
    p = e / wave_sum32(e);
    __syncthreads();
  }

  pv[L] = p;
  __syncthreads();

  // u = Z^T p + xn  (lane owns d = 4L..4L+3)
  float4 acc = xs;
#pragma unroll
  for (int j = 0; j < 8; ++j) {
    float4 pj = *(const float4*)&pv[4 * j];
    float4 z0 = *(const float4*)&zt[4 * j + 0][c4];
    float4 z1 = *(const float4*)&zt[4 * j + 1][c4];
    float4 z2 = *(const float4*)&zt[4 * j + 2][c4];
    float4 z3 = *(const float4*)&zt[4 * j + 3][c4];
    acc.x += pj.x * z0.x + pj.y * z1.x + pj.z * z2.x + pj.w * z3.x;
    acc.y += pj.x * z0.y + pj.y * z1.y + pj.z * z2.y + pj.w * z3.y;
    acc.z += pj.x * z0.z + pj.y * z1.z + pj.z * z2.z + pj.w * z3.z;
    acc.w += pj.x * z0.w + pj.y * z1.w + pj.z * z2.w + pj.w * z3.w;
  }
  ((float4*)out)[(size_t)node * (D_DIM / 4) + L] = acc;
}

extern "C" void kernel_launch(void* const* d_in, const int* in_sizes, int n_in,
                              void* d_out, int out_size, void* d_ws, size_t ws_size,
                              hipStream_t stream) {
  (void)n_in; (void)out_size; (void)ws_size;
  const float* x    = (const float*)d_in[0];
  const int*   x_nb = (const int*)d_in[1];
  float*       out  = (float*)d_out;
  int n = in_sizes[0] / D_DIM;               // 50000
  float* xn = (float*)d_ws;                  // n * 128 floats of scratch

  l2norm_rows<<<dim3((n + 7) / 8), dim3(256), 0, stream>>>(x, xn, n);
  capsule_route<<<dim3(n), dim3(32), 0, stream>>>(xn, x_nb, out, n);
}